// HybridGCN_75505525063863
// MI455X (gfx1250) — compile-verified
//
#include <hip/hip_runtime.h>
#include <hip/hip_bf16.h>

#define N_NODES  100000
#define N_EDGES  1600000
#define N_GRAPHS 512
#define GCN_IN   64
#define HID      128
#define RAD      45
#define FUSED    (HID + RAD)   // 173
#define EPS      1e-5f
#define PADF     4             // LDS row pad (floats) -> conflict-free fragment reads

typedef float v2f __attribute__((ext_vector_type(2)));
typedef float v8f __attribute__((ext_vector_type(8)));
typedef int   v4i __attribute__((vector_size(16)));

#if __has_builtin(__builtin_amdgcn_global_load_async_to_lds_b128)
#define HAVE_ASYNC_LDS 1
typedef __attribute__((address_space(1))) v4i* as1_v4i_p;   // prints as "__device__ v4i*"
typedef __attribute__((address_space(3))) v4i* as3_v4i_p;   // prints as "__shared__ v4i*"
#else
#define HAVE_ASYNC_LDS 0
#endif

// ---------------- utility kernels ----------------

__global__ void zero_f32(float* __restrict__ p, long n) {
    long i = (long)blockIdx.x * blockDim.x + threadIdx.x;
    if (i < n) p[i] = 0.f;
}

__global__ void deg_count(const int* __restrict__ dst, float* __restrict__ deg, int nE) {
    int e = blockIdx.x * blockDim.x + threadIdx.x;
    if (e < nE) atomicAdd(&deg[dst[e]], 1.f);
}

__global__ void deg_recip(float* __restrict__ deg, int n) {
    int i = blockIdx.x * blockDim.x + threadIdx.x;
    if (i < n) deg[i] = 1.f / fmaxf(deg[i], 1.f);
}

// scatter-add of gathered source features into agg[dst]
__global__ void agg_scatter(const float* __restrict__ h, const int* __restrict__ src,
                            const int* __restrict__ dst, float* __restrict__ agg,
                            int nE, int F) {
    long idx = (long)blockIdx.x * blockDim.x + threadIdx.x;
    int chunks = F >> 2;
    long total = (long)nE * chunks;
    if (idx >= total) return;
    int e = (int)(idx / chunks);
    int c = (int)(idx % chunks) * 4;
    int s = src[e], d = dst[e];
    const float4 v = *(const float4*)(h + (long)s * F + c);
    float* o = agg + (long)d * F + c;
    atomicAdd(o + 0, v.x);
    atomicAdd(o + 1, v.y);
    atomicAdd(o + 2, v.z);
    atomicAdd(o + 3, v.w);
}

// ---------------- fused SAGE linear: WMMA f32 16x16x4, LDS-staged A ----------------
// out = ReLU( BN( (deg_inv*agg)@Wl + x@Wr + bl ) ), HID = 128 columns.
// Block = 512 threads = 16 waves covering 32 rows x 128 cols:
//   wave = rt(0/1) x ct(0..7) -> 16x16 output tile at (rt*16, ct*16).
// A tiles (agg pre-scaled by deg_inv, and x) staged in LDS once per block;
// x tile uses GLOBAL_LOAD_ASYNC_TO_LDS_B128 (ASYNCcnt-tracked DMA).
__global__ void __launch_bounds__(512)
sage_wmma(const float* __restrict__ agg, const float* __restrict__ xin,
          const float* __restrict__ deg_inv,
          const float* __restrict__ Wl, const float* __restrict__ Wr,
          const float* __restrict__ bl,
          const float* __restrict__ bng, const float* __restrict__ bnb,
          const float* __restrict__ bnm, const float* __restrict__ bnv,
          float* __restrict__ out, int Fin) {
    __shared__ float sA[32 * (HID + PADF)];
    __shared__ float sX[32 * (HID + PADF)];

    const int FinP = Fin + PADF;
    const int tid  = threadIdx.x;
    const int lane = tid & 31;
    const int wave = tid >> 5;          // 0..15
    const int rt   = wave >> 3;         // row tile 0/1
    const int ct   = wave & 7;          // col tile 0..7
    const int rowb = blockIdx.x * 32;
    const int kh   = lane >> 4;         // K-pair selector
    const int l16  = lane & 15;

    // ---- stage 32 x Fin tiles into LDS ----
    const int cpr    = Fin >> 2;        // float4 chunks per row
    const int chunks = 32 * cpr;
    for (int c = tid; c < chunks; c += 512) {
        const int r  = c / cpr;
        const int fc = (c % cpr) << 2;
        // agg tile, pre-scaled by deg_inv (fold the mean divide into the stage)
        const float  dv = deg_inv[rowb + r];
        const float4 va = *(const float4*)(agg + (long)(rowb + r) * Fin + fc);
        float4 vs;
        vs.x = va.x * dv; vs.y = va.y * dv; vs.z = va.z * dv; vs.w = va.w * dv;
        *(float4*)(sA + r * FinP + fc) = vs;
        // x tile: async DMA memory -> LDS when available, else load + ds_store
#if HAVE_ASYNC_LDS
        __builtin_amdgcn_global_load_async_to_lds_b128(
            (as1_v4i_p)(xin + (long)(rowb + r) * Fin + fc),
            (as3_v4i_p)(sX + r * FinP + fc), 0, 0);
#else
        *(float4*)(sX + r * FinP + fc) =
            *(const float4*)(xin + (long)(rowb + r) * Fin + fc);
#endif
    }
#if HAVE_ASYNC_LDS
#if __has_builtin(__builtin_amdgcn_s_wait_asynccnt)
    __builtin_amdgcn_s_wait_asynccnt(0);
#else
    asm volatile("s_wait_asynccnt 0x0" ::: "memory");
#endif
#endif
    __syncthreads();

    // ---- WMMA K-loop ----
    const int    col  = ct * 16 + l16;
    const float* rowA = sA + (rt * 16 + l16) * FinP;
    const float* rowX = sX + (rt * 16 + l16) * FinP;

    v8f acc = {};
    for (int k = 0; k < Fin; k += 4) {
        const int kk = k + kh * 2;
        // (deg_inv*agg) @ Wl
        const v2f a = *(const v2f*)(rowA + kk);
        const float* bp = Wl + (long)kk * HID + col;
        v2f b; b.x = bp[0]; b.y = bp[HID];
        acc = __builtin_amdgcn_wmma_f32_16x16x4_f32(false, a, false, b,
                                                    (short)0, acc, false, false);
        // x @ Wr
        const v2f a2 = *(const v2f*)(rowX + kk);
        const float* bp2 = Wr + (long)kk * HID + col;
        v2f b2; b2.x = bp2[0]; b2.y = bp2[HID];
        acc = __builtin_amdgcn_wmma_f32_16x16x4_f32(false, a2, false, b2,
                                                    (short)0, acc, false, false);
    }

    // ---- epilogue: fold bias into BN, then ReLU ----
    const float s  = bng[col] * rsqrtf(bnv[col] + EPS);
    const float sh = bnb[col] + (bl[col] - bnm[col]) * s;
    const int   rbase = rowb + rt * 16 + 8 * kh;
    #pragma unroll
    for (int v = 0; v < 8; ++v) {
        float val = acc[v] * s + sh;
        val = fmaxf(val, 0.f);
        out[(long)(rbase + v) * HID + col] = val;
    }
}

// ---------------- pooling / fusion / head ----------------

__global__ void pool_scatter(const float* __restrict__ h, const int* __restrict__ batch,
                             float* __restrict__ gsum, float* __restrict__ gcnt) {
    const int chunks = HID / 4;
    long idx = (long)blockIdx.x * blockDim.x + threadIdx.x;
    if (idx >= (long)N_NODES * chunks) return;
    int node = (int)(idx / chunks);
    int c = (int)(idx % chunks) * 4;
    int g = batch[node];
    const float4 v = *(const float4*)(h + (long)node * HID + c);
    float* o = gsum + (long)g * HID + c;
    atomicAdd(o + 0, v.x);
    atomicAdd(o + 1, v.y);
    atomicAdd(o + 2, v.z);
    atomicAdd(o + 3, v.w);
    if (c == 0) atomicAdd(&gcnt[g], 1.f);
}

__global__ void fuse_kernel(const float* __restrict__ gsum, const float* __restrict__ gcnt,
                            const float* __restrict__ rad,
                            const float* __restrict__ rg, const float* __restrict__ rb,
                            const float* __restrict__ rm, const float* __restrict__ rv,
                            float* __restrict__ fused) {
    int idx = blockIdx.x * blockDim.x + threadIdx.x;
    if (idx >= N_GRAPHS * FUSED) return;
    int g = idx / FUSED, j = idx % FUSED;
    float v;
    if (j < HID) {
        v = gsum[(long)g * HID + j] / fmaxf(gcnt[g], 1.f);
    } else {
        int r = j - HID;
        v = (rad[(long)g * RAD + r] - rm[r]) * (rg[r] * rsqrtf(rv[r] + EPS)) + rb[r];
    }
    fused[idx] = v;
}

__global__ void linrelu(const float* __restrict__ in, const float* __restrict__ W,
                        const float* __restrict__ bias, float* __restrict__ out,
                        int M, int K, int N, int relu) {
    int idx = blockIdx.x * blockDim.x + threadIdx.x;
    if (idx >= M * N) return;
    int m = idx / N, n = idx % N;
    float s = bias[n];
    const float* ip = in + (long)m * K;
    for (int k = 0; k < K; ++k) s += ip[k] * W[(long)k * N + n];
    if (relu) s = fmaxf(s, 0.f);
    out[(long)m * N + n] = s;
}

// ---------------- host orchestration ----------------

static inline int grid1(long n, int b) { return (int)((n + b - 1) / b); }

extern "C" void kernel_launch(void* const* d_in, const int* in_sizes, int n_in,
                              void* d_out, int out_size, void* d_ws, size_t ws_size,
                              hipStream_t stream) {
    const float* x     = (const float*)d_in[0];
    const int*   eidx  = (const int*)d_in[1];
    const int*   batch = (const int*)d_in[2];
    const float* radio = (const float*)d_in[3];
    const float* Wl0 = (const float*)d_in[4];
    const float* bl0 = (const float*)d_in[5];
    const float* Wr0 = (const float*)d_in[6];
    const float* Wl1 = (const float*)d_in[7];
    const float* bl1 = (const float*)d_in[8];
    const float* Wr1 = (const float*)d_in[9];
    const float* Wl2 = (const float*)d_in[10];
    const float* bl2 = (const float*)d_in[11];
    const float* Wr2 = (const float*)d_in[12];
    const float* bn_g = (const float*)d_in[13];
    const float* bn_b = (const float*)d_in[14];
    const float* bn_m = (const float*)d_in[15];
    const float* bn_v = (const float*)d_in[16];
    const float* rad_g = (const float*)d_in[17];
    const float* rad_b = (const float*)d_in[18];
    const float* rad_m = (const float*)d_in[19];
    const float* rad_v = (const float*)d_in[20];
    const float* c1W = (const float*)d_in[21];
    const float* c1b = (const float*)d_in[22];
    const float* c2W = (const float*)d_in[23];
    const float* c2b = (const float*)d_in[24];
    const float* c3W = (const float*)d_in[25];
    const float* c3b = (const float*)d_in[26];
    const float* eW  = (const float*)d_in[27];
    const float* eb  = (const float*)d_in[28];

    const int* src = eidx;
    const int* dst = eidx + N_EDGES;

    // workspace carve-up
    float* ws    = (float*)d_ws;
    float* deg   = ws;                              // N_NODES (becomes deg_inv)
    float* agg   = deg  + N_NODES;                  // N_NODES*HID
    float* hA    = agg  + (long)N_NODES * HID;      // N_NODES*HID
    float* hB    = hA   + (long)N_NODES * HID;      // N_NODES*HID
    float* gsum  = hB   + (long)N_NODES * HID;      // N_GRAPHS*HID
    float* gcnt  = gsum + N_GRAPHS * HID;           // N_GRAPHS
    float* fused = gcnt + N_GRAPHS;                 // N_GRAPHS*FUSED
    float* hh1   = fused + N_GRAPHS * FUSED;        // N_GRAPHS*64
    float* hh2   = hh1  + N_GRAPHS * 64;            // N_GRAPHS*32

    // outputs (concatenated in return order)
    float* logits  = (float*)d_out;                 // 512*2
    float* emb     = logits + N_GRAPHS * 2;         // 512*173
    float* nodeemb = emb + N_GRAPHS * FUSED;        // 100000*128

    const int B = 256;

    // degree -> deg_inv
    zero_f32<<<grid1(N_NODES, B), B, 0, stream>>>(deg, N_NODES);
    deg_count<<<grid1(N_EDGES, B), B, 0, stream>>>(dst, deg, N_EDGES);
    deg_recip<<<grid1(N_NODES, B), B, 0, stream>>>(deg, N_NODES);

    const int rowTiles = N_NODES / 32;  // 3125, exact

    // ---- layer 0 (Fin = 64) ----
    {
        long aggN = (long)N_NODES * GCN_IN;
        zero_f32<<<grid1(aggN, B), B, 0, stream>>>(agg, aggN);
        long work = (long)N_EDGES * (GCN_IN / 4);
        agg_scatter<<<grid1(work, B), B, 0, stream>>>(x, src, dst, agg, N_EDGES, GCN_IN);
        sage_wmma<<<rowTiles, 512, 0, stream>>>(agg, x, deg, Wl0, Wr0, bl0,
                                                bn_g + 0 * HID, bn_b + 0 * HID,
                                                bn_m + 0 * HID, bn_v + 0 * HID,
                                                hA, GCN_IN);
    }
    // ---- layer 1 (Fin = 128) ----
    {
        long aggN = (long)N_NODES * HID;
        zero_f32<<<grid1(aggN, B), B, 0, stream>>>(agg, aggN);
        long work = (long)N_EDGES * (HID / 4);
        agg_scatter<<<grid1(work, B), B, 0, stream>>>(hA, src, dst, agg, N_EDGES, HID);
        sage_wmma<<<rowTiles, 512, 0, stream>>>(agg, hA, deg, Wl1, Wr1, bl1,
                                                bn_g + 1 * HID, bn_b + 1 * HID,
                                                bn_m + 1 * HID, bn_v + 1 * HID,
                                                hB, HID);
    }
    // ---- layer 2 (Fin = 128) -> node_emb output region ----
    {
        long aggN = (long)N_NODES * HID;
        zero_f32<<<grid1(aggN, B), B, 0, stream>>>(agg, aggN);
        long work = (long)N_EDGES * (HID / 4);
        agg_scatter<<<grid1(work, B), B, 0, stream>>>(hB, src, dst, agg, N_EDGES, HID);
        sage_wmma<<<rowTiles, 512, 0, stream>>>(agg, hB, deg, Wl2, Wr2, bl2,
                                                bn_g + 2 * HID, bn_b + 2 * HID,
                                                bn_m + 2 * HID, bn_v + 2 * HID,
                                                nodeemb, HID);
    }

    // ---- global mean pool + radiomics fuse ----
    long poolN = (long)N_GRAPHS * HID + N_GRAPHS;
    zero_f32<<<grid1(poolN, B), B, 0, stream>>>(gsum, poolN);  // gsum + gcnt contiguous
    long poolWork = (long)N_NODES * (HID / 4);
    pool_scatter<<<grid1(poolWork, B), B, 0, stream>>>(nodeemb, batch, gsum, gcnt);
    fuse_kernel<<<grid1((long)N_GRAPHS * FUSED, B), B, 0, stream>>>(
        gsum, gcnt, radio, rad_g, rad_b, rad_m, rad_v, fused);

    // ---- head MLP + embedding ----
    linrelu<<<grid1((long)N_GRAPHS * 64, B), B, 0, stream>>>(fused, c1W, c1b, hh1,
                                                             N_GRAPHS, FUSED, 64, 1);
    linrelu<<<grid1((long)N_GRAPHS * 32, B), B, 0, stream>>>(hh1, c2W, c2b, hh2,
                                                             N_GRAPHS, 64, 32, 1);
    linrelu<<<grid1((long)N_GRAPHS * 2, B), B, 0, stream>>>(hh2, c3W, c3b, logits,
                                                            N_GRAPHS, 32, 2, 0);
    linrelu<<<grid1((long)N_GRAPHS * FUSED, B), B, 0, stream>>>(fused, eW, eb, emb,
                                                                N_GRAPHS, FUSED, FUSED, 0);
}